// VN_PointNet_AM_44478681317793
// MI455X (gfx1250) — compile-verified
//
#include <hip/hip_runtime.h>
#include <hip/hip_bf16.h>
#include <math.h>

#define EPSV    1e-6f
#define BN_EPSV 1e-5f
#define BATCH 16
#define NPTS  2048
#define KNN   20
#define C1    21
#define C2    42
#define C2P   44      // padded K for WMMA (k0+1 reaches 43)
#define FCH   341
#define FPAD  352     // padded channel count (22 tiles of 16)

typedef float v2f __attribute__((ext_vector_type(2)));
typedef float v8f __attribute__((ext_vector_type(8)));

__device__ inline unsigned fenc(float f) {
    unsigned u = __float_as_uint(f);
    return (u & 0x80000000u) ? ~u : (u | 0x80000000u);
}
__device__ inline float fdec(unsigned k) {
    unsigned u = (k & 0x80000000u) ? (k ^ 0x80000000u) : ~k;
    return __uint_as_float(u);
}

// ---------------------------------------------------------------- kNN top-20
__global__ __launch_bounds__(128) void k_topk(const float* __restrict__ x,
                                              int* __restrict__ idx) {
    __shared__ float sx[3 * NPTS];
    __shared__ float sxx[NPTS];
    int b = blockIdx.y;
    for (int t = threadIdx.x; t < 3 * NPTS; t += 128) sx[t] = x[b * 3 * NPTS + t];
    __syncthreads();
    for (int t = threadIdx.x; t < NPTS; t += 128) {
        float a0 = sx[t], a1 = sx[NPTS + t], a2 = sx[2 * NPTS + t];
        sxx[t] = a0 * a0 + a1 * a1 + a2 * a2;
    }
    __syncthreads();
    int n = blockIdx.x * 128 + threadIdx.x;
    float c0 = sx[n], c1 = sx[NPTS + n], c2 = sx[2 * NPTS + n];
    float xxn = sxx[n];
    float vals[KNN]; int ids[KNN];
#pragma unroll
    for (int j = 0; j < KNN; ++j) { vals[j] = -INFINITY; ids[j] = 0; }
    for (int m = 0; m < NPTS; ++m) {
        float dot = c0 * sx[m] + c1 * sx[NPTS + m] + c2 * sx[2 * NPTS + m];
        float d = -(xxn - 2.0f * dot + sxx[m]);
        if (d > vals[KNN - 1]) {
            float cv = d; int ci = m;
#pragma unroll
            for (int j = 0; j < KNN; ++j) {
                bool sw = cv > vals[j];
                float tv = vals[j]; int ti = ids[j];
                if (sw) { vals[j] = cv; ids[j] = ci; cv = tv; ci = ti; }
            }
        }
    }
    int base = (b * NPTS + n) * KNN;
#pragma unroll
    for (int j = 0; j < KNN; ++j) idx[base + j] = ids[j];
}

// ---------------------------------------------------------------- utilities
__global__ void k_zero2(float* a, float* b, int n) {
    int i = blockIdx.x * blockDim.x + threadIdx.x;
    if (i < n) { a[i] = 0.f; b[i] = 0.f; }
}
__global__ void k_headinit(unsigned* mk, float* sb, int n) {
    int i = blockIdx.x * blockDim.x + threadIdx.x;
    if (i < n) { mk[i] = fenc(-INFINITY); sb[i] = 0.f; }
}
__global__ void k_finalize(const float* sum, const float* ssq,
                           float* mean, float* istd, int C, float cnt) {
    int i = blockIdx.x * blockDim.x + threadIdx.x;
    if (i < C) {
        float m = sum[i] / cnt;
        float v = ssq[i] / cnt - m * m;
        mean[i] = m;
        istd[i] = 1.0f / sqrtf(v + BN_EPSV);
    }
}
// zero-padded repack of W3 (341x42) into (352x44)
__global__ void k_packW3(const float* __restrict__ W3, float* __restrict__ W3p) {
    int i = blockIdx.x * blockDim.x + threadIdx.x;
    if (i >= FPAD * C2P) return;
    int f = i / C2P, c = i % C2P;
    W3p[i] = (f < FCH && c < C2) ? W3[f * C2 + c] : 0.f;
}
// zero the two pad channels (42,43) of the padded h3 tensor
__global__ void k_h3pad(float* __restrict__ h3p) {
    int i = blockIdx.x * blockDim.x + threadIdx.x;
    if (i >= BATCH * 2 * 3 * NPTS) return;
    int b = i / (2 * 3 * NPTS), off = i % (2 * 3 * NPTS);
    h3p[(b * C2P + C2) * 3 * NPTS + off] = 0.f;
}

// ---------------------------------------------------------------- edge layer (3ch feat -> 21)
__global__ __launch_bounds__(256) void k_p_stats(const float* __restrict__ x,
                                                 const int* __restrict__ idx,
                                                 const float* __restrict__ Wf,
                                                 float* __restrict__ sum,
                                                 float* __restrict__ ssq) {
    __shared__ float ls[C1], lq[C1];
    if (threadIdx.x < C1) { ls[threadIdx.x] = 0.f; lq[threadIdx.x] = 0.f; }
    __syncthreads();
    int gid = blockIdx.x * 256 + threadIdx.x;
    if (gid < BATCH * NPTS * KNN) {
        int b = gid / (NPTS * KNN);
        int r = gid % (NPTS * KNN);
        int n = r / KNN, k = r % KNN;
        const float* xb = x + b * 3 * NPTS;
        float cx = xb[n], cy = xb[NPTS + n], cz = xb[2 * NPTS + n];
        int j = idx[(b * NPTS + n) * KNN + k];
        float nx = xb[j], ny = xb[NPTS + j], nz = xb[2 * NPTS + j];
        float e0 = nx - cx, e1 = ny - cy, e2 = nz - cz;
        float r0 = ny * cz - nz * cy, r1 = nz * cx - nx * cz, r2 = nx * cy - ny * cx;
#pragma unroll 1
        for (int o = 0; o < C1; ++o) {
            float w0 = Wf[o * 3], w1 = Wf[o * 3 + 1], w2 = Wf[o * 3 + 2];
            float p0 = w0 * e0 + w1 * cx + w2 * r0;
            float p1 = w0 * e1 + w1 * cy + w2 * r1;
            float p2 = w0 * e2 + w1 * cz + w2 * r2;
            float nrm = sqrtf(p0 * p0 + p1 * p1 + p2 * p2) + EPSV;
            atomicAdd(&ls[o], nrm);
            atomicAdd(&lq[o], nrm * nrm);
        }
    }
    __syncthreads();
    if (threadIdx.x < C1) {
        atomicAdd(&sum[threadIdx.x], ls[threadIdx.x]);
        atomicAdd(&ssq[threadIdx.x], lq[threadIdx.x]);
    }
}

__global__ __launch_bounds__(128) void k_p_apply(const float* __restrict__ x,
                                                 const int* __restrict__ idx,
                                                 const float* __restrict__ Wf,
                                                 const float* __restrict__ Wd,
                                                 const float* __restrict__ g,
                                                 const float* __restrict__ bb,
                                                 const float* __restrict__ mean,
                                                 const float* __restrict__ istd,
                                                 float* __restrict__ hout) {
    int gid = blockIdx.x * 128 + threadIdx.x;
    if (gid >= BATCH * NPTS) return;
    int b = gid / NPTS, n = gid % NPTS;
    const float* xb = x + b * 3 * NPTS;
    float cx = xb[n], cy = xb[NPTS + n], cz = xb[2 * NPTS + n];
    float acc[C1][3];
#pragma unroll
    for (int o = 0; o < C1; ++o) { acc[o][0] = 0.f; acc[o][1] = 0.f; acc[o][2] = 0.f; }
    for (int k = 0; k < KNN; ++k) {
        int j = idx[(b * NPTS + n) * KNN + k];
        float nx = xb[j], ny = xb[NPTS + j], nz = xb[2 * NPTS + j];
        float e0 = nx - cx, e1 = ny - cy, e2 = nz - cz;
        float r0 = ny * cz - nz * cy, r1 = nz * cx - nx * cz, r2 = nx * cy - ny * cx;
#pragma unroll 1
        for (int o = 0; o < C1; ++o) {
            float wf0 = Wf[o * 3], wf1 = Wf[o * 3 + 1], wf2 = Wf[o * 3 + 2];
            float p0 = wf0 * e0 + wf1 * cx + wf2 * r0;
            float p1 = wf0 * e1 + wf1 * cy + wf2 * r1;
            float p2 = wf0 * e2 + wf1 * cz + wf2 * r2;
            float wd0 = Wd[o * 3], wd1 = Wd[o * 3 + 1], wd2 = Wd[o * 3 + 2];
            float d0 = wd0 * e0 + wd1 * cx + wd2 * r0;
            float d1 = wd0 * e1 + wd1 * cy + wd2 * r1;
            float d2 = wd0 * e2 + wd1 * cz + wd2 * r2;
            float nrm = sqrtf(p0 * p0 + p1 * p1 + p2 * p2) + EPSV;
            float s = (g[o] * (nrm - mean[o]) * istd[o] + bb[o]) / nrm;
            p0 *= s; p1 *= s; p2 *= s;
            float dt = p0 * d0 + p1 * d1 + p2 * d2;
            if (dt < 0.f) {
                float f = dt / (d0 * d0 + d1 * d1 + d2 * d2 + EPSV);
                p0 -= f * d0; p1 -= f * d1; p2 -= f * d2;
            }
            acc[o][0] += p0; acc[o][1] += p1; acc[o][2] += p2;
        }
    }
    const float inv = 1.0f / KNN;
#pragma unroll
    for (int o = 0; o < C1; ++o) {
        hout[((b * C1 + o) * 3 + 0) * NPTS + n] = acc[o][0] * inv;
        hout[((b * C1 + o) * 3 + 1) * NPTS + n] = acc[o][1] * inv;
        hout[((b * C1 + o) * 3 + 2) * NPTS + n] = acc[o][2] * inv;
    }
}

// ---------------------------------------------------------------- VN blocks with Cin=21
__global__ __launch_bounds__(128) void k_lin_stats(const float* __restrict__ hin,
                                                   const float* __restrict__ W,
                                                   int Cout,
                                                   float* __restrict__ sum,
                                                   float* __restrict__ ssq) {
    __shared__ float ls[C2], lq[C2];
    for (int t = threadIdx.x; t < Cout; t += 128) { ls[t] = 0.f; lq[t] = 0.f; }
    __syncthreads();
    int gid = blockIdx.x * 128 + threadIdx.x;
    int b = gid / NPTS, n = gid % NPTS;
    float hv[C1][3];
#pragma unroll
    for (int c = 0; c < C1; ++c) {
        hv[c][0] = hin[((b * C1 + c) * 3 + 0) * NPTS + n];
        hv[c][1] = hin[((b * C1 + c) * 3 + 1) * NPTS + n];
        hv[c][2] = hin[((b * C1 + c) * 3 + 2) * NPTS + n];
    }
    for (int o = 0; o < Cout; ++o) {
        float z0 = 0.f, z1 = 0.f, z2 = 0.f;
#pragma unroll
        for (int c = 0; c < C1; ++c) {
            float w = W[o * C1 + c];
            z0 += w * hv[c][0]; z1 += w * hv[c][1]; z2 += w * hv[c][2];
        }
        float nrm = sqrtf(z0 * z0 + z1 * z1 + z2 * z2) + EPSV;
        atomicAdd(&ls[o], nrm);
        atomicAdd(&lq[o], nrm * nrm);
    }
    __syncthreads();
    for (int t = threadIdx.x; t < Cout; t += 128) {
        atomicAdd(&sum[t], ls[t]);
        atomicAdd(&ssq[t], lq[t]);
    }
}

// Cstride: channel stride of hout (21 for h2, 44 for zero-padded h3)
__global__ __launch_bounds__(128) void k_lin_apply(const float* __restrict__ hin,
                                                   const float* __restrict__ Wf,
                                                   const float* __restrict__ Wd,
                                                   const float* __restrict__ g,
                                                   const float* __restrict__ bb,
                                                   const float* __restrict__ mean,
                                                   const float* __restrict__ istd,
                                                   int Cout, int Cstride,
                                                   float* __restrict__ hout) {
    int gid = blockIdx.x * 128 + threadIdx.x;
    int b = gid / NPTS, n = gid % NPTS;
    float hv[C1][3];
#pragma unroll
    for (int c = 0; c < C1; ++c) {
        hv[c][0] = hin[((b * C1 + c) * 3 + 0) * NPTS + n];
        hv[c][1] = hin[((b * C1 + c) * 3 + 1) * NPTS + n];
        hv[c][2] = hin[((b * C1 + c) * 3 + 2) * NPTS + n];
    }
    for (int o = 0; o < Cout; ++o) {
        float p0 = 0.f, p1 = 0.f, p2 = 0.f, d0 = 0.f, d1 = 0.f, d2 = 0.f;
#pragma unroll
        for (int c = 0; c < C1; ++c) {
            float wf = Wf[o * C1 + c], wd = Wd[o * C1 + c];
            p0 += wf * hv[c][0]; p1 += wf * hv[c][1]; p2 += wf * hv[c][2];
            d0 += wd * hv[c][0]; d1 += wd * hv[c][1]; d2 += wd * hv[c][2];
        }
        float nrm = sqrtf(p0 * p0 + p1 * p1 + p2 * p2) + EPSV;
        float s = (g[o] * (nrm - mean[o]) * istd[o] + bb[o]) / nrm;
        p0 *= s; p1 *= s; p2 *= s;
        float dt = p0 * d0 + p1 * d1 + p2 * d2;
        if (dt < 0.f) {
            float f = dt / (d0 * d0 + d1 * d1 + d2 * d2 + EPSV);
            p0 -= f * d0; p1 -= f * d1; p2 -= f * d2;
        }
        hout[((b * Cstride + o) * 3 + 0) * NPTS + n] = p0;
        hout[((b * Cstride + o) * 3 + 1) * NPTS + n] = p1;
        hout[((b * Cstride + o) * 3 + 2) * NPTS + n] = p2;
    }
}

// ---------------------------------------------------------------- W3 GEMM via WMMA f32 16x16x4
// Operands are zero-padded (W3p: 352x44, h3p: 44-channel stride) -> branch-free fragment loads.
// A tile: W3p (M=channel f, K=c).  A layout: m = lane%16, k = vgpr + 2*(lane/16)
// B tile: h3p (K=c, N=point n).    B layout: n = lane%16, k = vgpr + 2*(lane/16)
// C/D:    lane l, vgpr r -> f = r + 8*(l/16), n = l%16
__device__ inline void w3_tiles(const float* __restrict__ h3p,
                                const float* __restrict__ W3p,
                                int b, int fb, int nb, int lane, v8f acc[3]) {
    int ln = lane & 15, lh = lane >> 4;
    int fr = fb + ln;
#pragma unroll
    for (int kc = 0; kc < 11; ++kc) {
        int k0 = 4 * kc + 2 * lh;                       // even -> 8B aligned
        v2f a = *(const v2f*)&W3p[fr * C2P + k0];
        const float* hb = h3p + (size_t)(b * C2P + k0) * 3 * NPTS + nb + ln;
#pragma unroll
        for (int v = 0; v < 3; ++v) {
            v2f bm;
            bm.x = hb[(size_t)v * NPTS];                // channel k0,   vec v
            bm.y = hb[(size_t)(3 + v) * NPTS];          // channel k0+1, vec v
            acc[v] = __builtin_amdgcn_wmma_f32_16x16x4_f32(
                false, a, false, bm, (short)0, acc[v], false, false);
        }
    }
}

__global__ __launch_bounds__(32) void k_w3_stats(const float* __restrict__ h3p,
                                                 const float* __restrict__ W3p,
                                                 float* __restrict__ sum,
                                                 float* __restrict__ ssq) {
    __shared__ float ls[16], lq[16];
    int lane = threadIdx.x;
    if (lane < 16) { ls[lane] = 0.f; lq[lane] = 0.f; }
    __syncthreads();
    int nb = blockIdx.x * 16, fb = blockIdx.y * 16, b = blockIdx.z;
    v8f acc[3] = {};
    w3_tiles(h3p, W3p, b, fb, nb, lane, acc);
    int lh = lane >> 4;
#pragma unroll
    for (int r = 0; r < 8; ++r) {
        float z0 = acc[0][r], z1 = acc[1][r], z2 = acc[2][r];
        float nrm = sqrtf(z0 * z0 + z1 * z1 + z2 * z2) + EPSV;
        atomicAdd(&ls[r + 8 * lh], nrm);
        atomicAdd(&lq[r + 8 * lh], nrm * nrm);
    }
    __syncthreads();
    if (lane < 16) {
        int f = fb + lane;
        if (f < FCH) { atomicAdd(&sum[f], ls[lane]); atomicAdd(&ssq[f], lq[lane]); }
    }
}

__global__ __launch_bounds__(32) void k_w3_apply(const float* __restrict__ h3p,
                                                 const float* __restrict__ W3p,
                                                 const float* __restrict__ g,
                                                 const float* __restrict__ bb,
                                                 const float* __restrict__ mean,
                                                 const float* __restrict__ istd,
                                                 unsigned* __restrict__ maxk,
                                                 float* __restrict__ sumb) {
    int lane = threadIdx.x;
    int nb = blockIdx.x * 16, fb = blockIdx.y * 16, b = blockIdx.z;
    v8f acc[3] = {};
    w3_tiles(h3p, W3p, b, fb, nb, lane, acc);
    int lh = lane >> 4, ln = lane & 15;
#pragma unroll
    for (int r = 0; r < 8; ++r) {
        int f = fb + r + 8 * lh;
        bool valid = f < FCH;
        float z0 = acc[0][r], z1 = acc[1][r], z2 = acc[2][r];
        float nrm = sqrtf(z0 * z0 + z1 * z1 + z2 * z2) + EPSV;
        float s = 0.f;
        if (valid) s = (g[f] * (nrm - mean[f]) * istd[f] + bb[f]) / nrm;
        float h0 = z0 * s, h1 = z1 * s, h2 = z2 * s;
        float m0 = h0, m1 = h1, m2 = h2, s0 = h0, s1 = h1, s2 = h2;
#pragma unroll
        for (int off = 1; off < 16; off <<= 1) {
            m0 = fmaxf(m0, __shfl_xor(m0, off));
            m1 = fmaxf(m1, __shfl_xor(m1, off));
            m2 = fmaxf(m2, __shfl_xor(m2, off));
            s0 += __shfl_xor(s0, off);
            s1 += __shfl_xor(s1, off);
            s2 += __shfl_xor(s2, off);
        }
        if (ln == 0 && valid) {
            int base = (b * FCH + f) * 3;
            atomicMax(&maxk[base + 0], fenc(m0));
            atomicMax(&maxk[base + 1], fenc(m1));
            atomicMax(&maxk[base + 2], fenc(m2));
            atomicAdd(&sumb[base + 0], s0);
            atomicAdd(&sumb[base + 1], s1);
            atomicAdd(&sumb[base + 2], s2);
        }
    }
}

__global__ void k_head_combine(const unsigned* __restrict__ maxk,
                               const float* __restrict__ sumb,
                               float* __restrict__ hh) {
    int i = blockIdx.x * blockDim.x + threadIdx.x;
    if (i >= BATCH * 682 * 3) return;
    int v = i % 3;
    int c = (i / 3) % 682;
    int b = i / (682 * 3);
    float val;
    if (c < FCH) val = fdec(maxk[(b * FCH + c) * 3 + v]);
    else         val = sumb[(b * FCH + (c - FCH)) * 3 + v] * (1.0f / NPTS);
    hh[i] = val;
}

// ---------------------------------------------------------------- head: 682->341 x3 + Gram-Schmidt
__global__ __launch_bounds__(128) void k_lin3_bgs(const float* __restrict__ X,
                                                  const float* __restrict__ Wv,
                                                  const float* __restrict__ Wj,
                                                  const float* __restrict__ Wjb,
                                                  float* __restrict__ vout,
                                                  float* __restrict__ Rout,
                                                  float* __restrict__ invout) {
    int gid = blockIdx.x * 128 + threadIdx.x;
    if (gid >= BATCH * FCH) return;
    int b = gid / FCH, o = gid % FCH;
    float v0 = 0, v1 = 0, v2 = 0, j0 = 0, j1 = 0, j2 = 0, q0 = 0, q1 = 0, q2 = 0;
    for (int c = 0; c < 682; ++c) {
        float x0 = X[(b * 682 + c) * 3 + 0];
        float x1 = X[(b * 682 + c) * 3 + 1];
        float x2 = X[(b * 682 + c) * 3 + 2];
        float wv = Wv[o * 682 + c], wj = Wj[o * 682 + c], wq = Wjb[o * 682 + c];
        v0 += wv * x0; v1 += wv * x1; v2 += wv * x2;
        j0 += wj * x0; j1 += wj * x1; j2 += wj * x2;
        q0 += wq * x0; q1 += wq * x1; q2 += wq * x2;
    }
    float n1 = sqrtf(j0 * j0 + j1 * j1 + j2 * j2) + EPSV;
    float b10 = j0 / n1, b11 = j1 / n1, b12 = j2 / n1;
    float dp = b10 * q0 + b11 * q1 + b12 * q2;
    float a0 = q0 - dp * b10, a1 = q1 - dp * b11, a2 = q2 - dp * b12;
    float n2 = sqrtf(a0 * a0 + a1 * a1 + a2 * a2) + EPSV;
    float b20 = a0 / n2, b21 = a1 / n2, b22 = a2 / n2;
    float b30 = b11 * b22 - b12 * b21;
    float b31 = b12 * b20 - b10 * b22;
    float b32 = b10 * b21 - b11 * b20;
    int rb = gid * 9;
    Rout[rb + 0] = b10; Rout[rb + 1] = b20; Rout[rb + 2] = b30;
    Rout[rb + 3] = b11; Rout[rb + 4] = b21; Rout[rb + 5] = b31;
    Rout[rb + 6] = b12; Rout[rb + 7] = b22; Rout[rb + 8] = b32;
    invout[gid * 3 + 0] = b10 * v0 + b11 * v1 + b12 * v2;
    invout[gid * 3 + 1] = b20 * v0 + b21 * v1 + b22 * v2;
    invout[gid * 3 + 2] = b30 * v0 + b31 * v1 + b32 * v2;
    vout[gid * 3 + 0] = v0; vout[gid * 3 + 1] = v1; vout[gid * 3 + 2] = v2;
}

__global__ __launch_bounds__(128) void k_affine(const float* __restrict__ Wa,
                                                const float* __restrict__ vsrc,
                                                const float* __restrict__ Rbuf,
                                                const float* __restrict__ invbuf,
                                                float* __restrict__ out) {
    int gid = blockIdx.x * 128 + threadIdx.x;
    if (gid >= BATCH * FCH) return;
    int b = gid / FCH, o = gid % FCH;
    float z0 = 0, z1 = 0, z2 = 0;
    const float* w = Wa + o * FCH;
    const float* iv = invbuf + b * FCH * 3;
    for (int c = 0; c < FCH; ++c) {
        float ww = w[c];
        z0 += ww * iv[c * 3 + 0];
        z1 += ww * iv[c * 3 + 1];
        z2 += ww * iv[c * 3 + 2];
    }
    const float* R = Rbuf + gid * 9;
    float y0 = R[0] * z0 + R[1] * z1 + R[2] * z2;
    float y1 = R[3] * z0 + R[4] * z1 + R[5] * z2;
    float y2 = R[6] * z0 + R[7] * z1 + R[8] * z2;
    int ob = (b * 682 + o) * 3;
    out[ob + 0] = vsrc[gid * 3 + 0];
    out[ob + 1] = vsrc[gid * 3 + 1];
    out[ob + 2] = vsrc[gid * 3 + 2];
    int ob2 = (b * 682 + FCH + o) * 3;
    out[ob2 + 0] = y0; out[ob2 + 1] = y1; out[ob2 + 2] = y2;
}

// ---------------------------------------------------------------- launcher
extern "C" void kernel_launch(void* const* d_in, const int* in_sizes, int n_in,
                              void* d_out, int out_size, void* d_ws, size_t ws_size,
                              hipStream_t stream) {
    (void)in_sizes; (void)n_in; (void)out_size; (void)ws_size;
    const float* x    = (const float*)d_in[0];
    const float* Wp_f = (const float*)d_in[1];
    const float* Wp_d = (const float*)d_in[2];
    const float* gp   = (const float*)d_in[3];
    const float* bp   = (const float*)d_in[4];
    const float* W1f  = (const float*)d_in[5];
    const float* W1d  = (const float*)d_in[6];
    const float* g1   = (const float*)d_in[7];
    const float* b1   = (const float*)d_in[8];
    const float* W2f  = (const float*)d_in[9];
    const float* W2d  = (const float*)d_in[10];
    const float* g2   = (const float*)d_in[11];
    const float* b2   = (const float*)d_in[12];
    const float* W3   = (const float*)d_in[13];
    const float* g3   = (const float*)d_in[14];
    const float* b3   = (const float*)d_in[15];
    const float* Wl1  = (const float*)d_in[16];
    const float* Wl2  = (const float*)d_in[17];
    const float* Wl2b = (const float*)d_in[18];
    const float* Wl3  = (const float*)d_in[19];
    const float* Wl4  = (const float*)d_in[20];
    const float* Wl4b = (const float*)d_in[21];
    const float* Wa1  = (const float*)d_in[22];
    const float* Wa2  = (const float*)d_in[23];
    float* out = (float*)d_out;

    char* ws = (char*)d_ws;
    size_t off = 0;
    auto alloc = [&](size_t nfloats) -> float* {
        float* p = (float*)(ws + off);
        off += ((nfloats * 4 + 255) / 256) * 256;
        return p;
    };
    int*      idx   = (int*)alloc((size_t)BATCH * NPTS * KNN);
    float*    h1    = alloc((size_t)BATCH * C1 * 3 * NPTS);
    float*    h2    = alloc((size_t)BATCH * C1 * 3 * NPTS);
    float*    h3p   = alloc((size_t)BATCH * C2P * 3 * NPTS);  // padded channels
    float*    W3p   = alloc((size_t)FPAD * C2P);              // padded weights
    float*    sum   = alloc(FPAD);
    float*    ssq   = alloc(FPAD);
    float*    mean  = alloc(FPAD);
    float*    istd  = alloc(FPAD);
    unsigned* maxk  = (unsigned*)alloc((size_t)BATCH * FCH * 3);
    float*    sumb  = alloc((size_t)BATCH * FCH * 3);
    float*    hh    = alloc((size_t)BATCH * 682 * 3);
    float*    vbuf  = alloc((size_t)BATCH * FCH * 3);
    float*    Rb    = alloc((size_t)BATCH * FCH * 9);
    float*    invb  = alloc((size_t)BATCH * FCH * 3);
    float*    comb  = alloc((size_t)BATCH * 682 * 3);
    float*    x2b   = alloc((size_t)BATCH * FCH * 3);
    float*    Rb2   = alloc((size_t)BATCH * FCH * 9);
    float*    invb2 = alloc((size_t)BATCH * FCH * 3);

    // 1) kNN graph
    k_topk<<<dim3(NPTS / 128, BATCH), 128, 0, stream>>>(x, idx);

    // 2) edge VN block (3 -> 21), mean over k
    k_zero2<<<2, 256, 0, stream>>>(sum, ssq, FPAD);
    k_p_stats<<<(BATCH * NPTS * KNN) / 256, 256, 0, stream>>>(x, idx, Wp_f, sum, ssq);
    k_finalize<<<2, 256, 0, stream>>>(sum, ssq, mean, istd, C1, (float)(BATCH * NPTS * KNN));
    k_p_apply<<<(BATCH * NPTS) / 128, 128, 0, stream>>>(x, idx, Wp_f, Wp_d, gp, bp, mean, istd, h1);

    // 3) VN block 21 -> 21
    k_zero2<<<2, 256, 0, stream>>>(sum, ssq, FPAD);
    k_lin_stats<<<(BATCH * NPTS) / 128, 128, 0, stream>>>(h1, W1f, C1, sum, ssq);
    k_finalize<<<2, 256, 0, stream>>>(sum, ssq, mean, istd, C1, (float)(BATCH * NPTS));
    k_lin_apply<<<(BATCH * NPTS) / 128, 128, 0, stream>>>(h1, W1f, W1d, g1, b1, mean, istd, C1, C1, h2);

    // 4) VN block 21 -> 42 (written with padded channel stride 44)
    k_zero2<<<2, 256, 0, stream>>>(sum, ssq, FPAD);
    k_lin_stats<<<(BATCH * NPTS) / 128, 128, 0, stream>>>(h2, W2f, C2, sum, ssq);
    k_finalize<<<2, 256, 0, stream>>>(sum, ssq, mean, istd, C2, (float)(BATCH * NPTS));
    k_lin_apply<<<(BATCH * NPTS) / 128, 128, 0, stream>>>(h2, W2f, W2d, g2, b2, mean, istd, C2, C2P, h3p);

    // 5) W3 (42 -> 341) via WMMA + BN + max/mean pooling over N
    k_packW3<<<(FPAD * C2P + 255) / 256, 256, 0, stream>>>(W3, W3p);
    k_h3pad<<<(BATCH * 2 * 3 * NPTS + 255) / 256, 256, 0, stream>>>(h3p);
    k_zero2<<<2, 256, 0, stream>>>(sum, ssq, FPAD);
    k_headinit<<<(BATCH * FCH * 3 + 255) / 256, 256, 0, stream>>>(maxk, sumb, BATCH * FCH * 3);
    k_w3_stats<<<dim3(NPTS / 16, FPAD / 16, BATCH), 32, 0, stream>>>(h3p, W3p, sum, ssq);
    k_finalize<<<2, 256, 0, stream>>>(sum, ssq, mean, istd, FCH, (float)(BATCH * NPTS));
    k_w3_apply<<<dim3(NPTS / 16, FPAD / 16, BATCH), 32, 0, stream>>>(h3p, W3p, g3, b3, mean, istd, maxk, sumb);
    k_head_combine<<<(BATCH * 682 * 3 + 255) / 256, 256, 0, stream>>>(maxk, sumb, hh);

    // 6) head stage 1
    k_lin3_bgs<<<(BATCH * FCH + 127) / 128, 128, 0, stream>>>(hh, Wl1, Wl2, Wl2b, vbuf, Rb, invb);
    k_affine<<<(BATCH * FCH + 127) / 128, 128, 0, stream>>>(Wa1, vbuf, Rb, invb, comb);

    // 7) head stage 2 -> output (16, 682, 3)
    k_lin3_bgs<<<(BATCH * FCH + 127) / 128, 128, 0, stream>>>(comb, Wl3, Wl4, Wl4b, x2b, Rb2, invb2);
    k_affine<<<(BATCH * FCH + 127) / 128, 128, 0, stream>>>(Wa2, x2b, Rb2, invb2, out);
}